// DeformableConvolutionBlock_63436666962356
// MI455X (gfx1250) — compile-verified
//
#include <hip/hip_runtime.h>
#include <math.h>

typedef __attribute__((ext_vector_type(2))) float v2f;
typedef __attribute__((ext_vector_type(4))) float v4f;
typedef __attribute__((ext_vector_type(8))) float v8f;
// 64-bit gather with 4-byte alignment (pair of adjacent pixels in a row).
typedef float f2u __attribute__((ext_vector_type(2), aligned(4)));

#define Bn   8
#define Cc   64
#define OCn  64
#define Hh   256
#define Ww   256
#define HWn  (Hh * Ww)
#define CHWn (Cc * HWn)

// Each wave32 handles 16 consecutive pixels of one image row.
// GEMM: out[oc, pix] = sum_c w_conv[oc,c] * sampled[c,pix]
// tiled as M=16 (oc) x N=16 (pixels) x K=64 (channels), K-step = 4
// using V_WMMA_F32_16X16X4_F32 (f32 reference -> f32 matrix path).
__launch_bounds__(256, 2)
__global__ void deform_conv1x1_wmma(const float* __restrict__ x,
                                    const float* __restrict__ w_conv,
                                    const float* __restrict__ w_off,
                                    const float* __restrict__ b_off,
                                    const float* __restrict__ w_mask,
                                    const float* __restrict__ b_mask,
                                    float* __restrict__ out)
{
    __shared__ float s_wconv[OCn * Cc];   // 16 KB
    __shared__ float s_woff[2 * Cc];
    __shared__ float s_wmask[Cc];
    // Per-wave transpose buffer for phase-1 partial sums:
    // [wave][pixel 0..15][cg 0..7][{dy,dx,m}] packed -> 24 floats per pixel row.
    __shared__ __attribute__((aligned(16))) float s_part[8][16][24];  // 12 KB

    // Cooperative stage of weights into LDS.
    for (int i = threadIdx.x; i < OCn * Cc; i += 256) s_wconv[i] = w_conv[i];
    if (threadIdx.x < 2 * Cc) s_woff[threadIdx.x] = w_off[threadIdx.x];
    if (threadIdx.x < Cc)     s_wmask[threadIdx.x] = w_mask[threadIdx.x];
    __syncthreads();

    const float boff_y = b_off[0];
    const float boff_x = b_off[1];
    const float bmsk   = b_mask[0];

    const int lane = threadIdx.x & 31;
    const int wave = threadIdx.x >> 5;
    const int p    = lane & 15;   // pixel within 16-wide tile (N index)
    const int half = lane >> 4;   // K half selector (ISA A/B lane striping)

    const long tileBase = ((long)blockIdx.x * 8 + wave) * 16; // flat pixel index
    const int b   = (int)(tileBase / HWn);
    const int hw0 = (int)(tileBase % HWn);
    const int hw  = hw0 + p;          // this lane's pixel (flat h*W+w)
    const int h   = hw >> 8;
    const int w   = hw & (Ww - 1);

    const float* __restrict__ xb = x + (size_t)b * CHWn;

    // ---- Phase 1: offset / mask 1x1 convs.
    // Lane l: pixel sub-block pg = l&3 (4 consecutive pixels, one b128 load),
    // channel group cg = l>>2 (8 channels). 8 x global_load_b128 per lane,
    // no duplicated addresses; reduce across cg via per-wave LDS transpose.
    {
        const int pg = lane & 3;
        const int cg = lane >> 2;
        const float* __restrict__ xpix = xb + hw0 + pg * 4;

        v4f a_dy = {0.f, 0.f, 0.f, 0.f};
        v4f a_dx = {0.f, 0.f, 0.f, 0.f};
        v4f a_m  = {0.f, 0.f, 0.f, 0.f};
        #pragma unroll
        for (int i = 0; i < 8; ++i) {
            const int c = cg * 8 + i;
            const v4f v = *(const v4f*)(xpix + (size_t)c * HWn);
            a_dy += v * s_woff[c];
            a_dx += v * s_woff[Cc + c];
            a_m  += v * s_wmask[c];
        }
        #pragma unroll
        for (int j = 0; j < 4; ++j) {
            float* row = &s_part[wave][pg * 4 + j][cg * 3];
            row[0] = a_dy[j];
            row[1] = a_dx[j];
            row[2] = a_m[j];
        }
    }
    // Same-wave DS ops are in-order (DScnt); barrier only pins compiler order.
    __builtin_amdgcn_wave_barrier();

    // Each lane sums the 8 channel-group partials for its own pixel p
    // (both lane-halves read the same row; replaces the old shfl_xor pass).
    float sdy, sdx, sm;
    {
        const v4f* q = (const v4f*)&s_part[wave][p][0];
        const v4f q0 = q[0], q1 = q[1], q2 = q[2], q3 = q[3], q4 = q[4], q5 = q[5];
        sdy = ((q0.x + q0.w) + (q1.z + q2.y)) + ((q3.x + q3.w) + (q4.z + q5.y));
        sdx = ((q0.y + q1.x) + (q1.w + q2.z)) + ((q3.y + q4.x) + (q4.w + q5.z));
        sm  = ((q0.z + q1.y) + (q2.x + q2.w)) + ((q3.z + q4.y) + (q5.x + q5.w));
    }

    const float py = (float)h + sdy + boff_y;
    const float px = (float)w + sdx + boff_x;
    const float mk = 1.f / (1.f + __expf(-(sm + bmsk)));

    // ---- Bilinear sampling parameters (validity folded into weights; EXEC stays full)
    const float y0f = floorf(py), x0f = floorf(px);
    const int   y0  = (int)y0f,   x0  = (int)x0f;
    const int   y1  = y0 + 1,     x1  = x0 + 1;
    const float wy1 = py - y0f,   wx1 = px - x0f;
    const float wy0 = 1.f - wy1,  wx0 = 1.f - wx1;

    const bool vy0 = (y0 >= 0) && (y0 < Hh);
    const bool vy1 = (y1 >= 0) && (y1 < Hh);
    const bool vx0 = (x0 >= 0) && (x0 < Ww);
    const bool vx1 = (x1 >= 0) && (x1 < Ww);

    const int y0c = min(max(y0, 0), Hh - 1);
    const int y1c = min(max(y1, 0), Hh - 1);
    const int x0c = min(max(x0, 0), Ww - 1);
    const int x1c = min(max(x1, 0), Ww - 1);

    const float w00 = (vy0 && vx0) ? wy0 * wx0 * mk : 0.f;
    const float w01 = (vy0 && vx1) ? wy0 * wx1 * mk : 0.f;
    const float w10 = (vy1 && vx0) ? wy1 * wx0 * mk : 0.f;
    const float w11 = (vy1 && vx1) ? wy1 * wx1 * mk : 0.f;

    // Pair-base for 64-bit gathers covering both x-columns of a corner row.
    // bx in [0, W-2]; x0c,x1c are always bx or bx+1, so one b64 gather per row
    // + two selects replaces two b32 gathers. Invalid corners have weight 0,
    // so the clamped in-bounds neighbor value is harmless.
    const int  bx = min(max(x0, 0), Ww - 2);
    const bool s0 = (x0c != bx);     // select high element for x0 column
    const bool s1 = (x1c != bx);     // select high element for x1 column
    const int  jr0 = y0c * Ww + bx;  // pair base, row y0
    const int  jr1 = y1c * Ww + bx;  // pair base, row y1

    // ---- Phase 2+3: fused sample + WMMA GEMM over K=64 in steps of 4
    v8f acc[4];
    #pragma unroll
    for (int m = 0; m < 4; ++m)
        #pragma unroll
        for (int r = 0; r < 8; ++r) acc[m][r] = 0.f;

    #pragma unroll 4
    for (int kk = 0; kk < 16; ++kk) {
        // B operand: 4x16 slab of sampled[c, pix].
        // Lane striping (32-bit A/B, 16x4): lanes 0-15 -> K = 4kk+{0,1},
        // lanes 16-31 -> K = 4kk+{2,3}; this lane covers channels c0, c0+1.
        const int c0 = kk * 4 + half * 2;
        const float* __restrict__ xc0 = xb + (size_t)c0 * HWn;
        const float* __restrict__ xc1 = xc0 + HWn;

        // 4 x 64-bit gathers (2 rows x 2 channels) instead of 8 x 32-bit.
        const f2u pr0c0 = *(const f2u*)(xc0 + jr0);
        const f2u pr1c0 = *(const f2u*)(xc0 + jr1);
        const f2u pr0c1 = *(const f2u*)(xc1 + jr0);
        const f2u pr1c1 = *(const f2u*)(xc1 + jr1);

        v2f Bm;
        {
            const float v00 = s0 ? pr0c0.y : pr0c0.x;  // (y0, x0)
            const float v01 = s1 ? pr0c0.y : pr0c0.x;  // (y0, x1)
            const float v10 = s0 ? pr1c0.y : pr1c0.x;  // (y1, x0)
            const float v11 = s1 ? pr1c0.y : pr1c0.x;  // (y1, x1)
            Bm.x = fmaf(w00, v00, fmaf(w01, v01, fmaf(w10, v10, w11 * v11)));
        }
        {
            const float v00 = s0 ? pr0c1.y : pr0c1.x;
            const float v01 = s1 ? pr0c1.y : pr0c1.x;
            const float v10 = s0 ? pr1c1.y : pr1c1.x;
            const float v11 = s1 ? pr1c1.y : pr1c1.x;
            Bm.y = fmaf(w00, v00, fmaf(w01, v01, fmaf(w10, v10, w11 * v11)));
        }

        #pragma unroll
        for (int m = 0; m < 4; ++m) {
            // A operand: w_conv tile, M = oc = 16m + (lane%16), K as above.
            const int oc = m * 16 + p;
            v2f Am;
            Am.x = s_wconv[oc * Cc + c0];
            Am.y = s_wconv[oc * Cc + c0 + 1];
            acc[m] = __builtin_amdgcn_wmma_f32_16x16x4_f32(
                false, Am, false, Bm, (short)0, acc[m], false, false);
        }
    }

    // ---- Store: D layout -> lane half selects M+8, reg r is M row.
    // Non-temporal: output is write-once; keep x resident in L2 instead.
    float* __restrict__ ob = out + (size_t)b * CHWn + hw;
    #pragma unroll
    for (int m = 0; m < 4; ++m) {
        #pragma unroll
        for (int r = 0; r < 8; ++r) {
            const int oc = m * 16 + half * 8 + r;
            __builtin_nontemporal_store(acc[m][r], ob + (size_t)oc * HWn);
        }
    }
}

extern "C" void kernel_launch(void* const* d_in, const int* in_sizes, int n_in,
                              void* d_out, int out_size, void* d_ws, size_t ws_size,
                              hipStream_t stream) {
    const float* x      = (const float*)d_in[0];
    const float* w_conv = (const float*)d_in[1];
    const float* w_off  = (const float*)d_in[2];
    const float* b_off  = (const float*)d_in[3];
    const float* w_mask = (const float*)d_in[4];
    const float* b_mask = (const float*)d_in[5];
    float* out = (float*)d_out;

    // 8*256*256 pixels / (8 waves * 16 pixels) = 4096 blocks of 256 threads.
    dim3 grid(4096), block(256);
    hipLaunchKernelGGL(deform_conv1x1_wmma, grid, block, 0, stream,
                       x, w_conv, w_off, b_off, w_mask, b_mask, out);
}